// MultiHeadAttention_80607946211923
// MI455X (gfx1250) — compile-verified
//
#include <hip/hip_runtime.h>
#include <hip/hip_bf16.h>

typedef __attribute__((ext_vector_type(2))) float v2f;
typedef __attribute__((ext_vector_type(8))) float v8f;
typedef __attribute__((ext_vector_type(4))) unsigned int u32x4;
typedef __attribute__((ext_vector_type(8))) int i32x8;
typedef __attribute__((ext_vector_type(4))) int i32x4;

#define B_   4
#define S_   2048
#define DIN_ 768
#define H_   12
#define DH_  64

// padded LDS row strides (stride % 64 == 4 -> 16 A-rows hit distinct banks)
#define XS_STRIDE  (DIN_ + 4)   // 772
#define QS_STRIDE  (DH_ + 4)    // 68
#define SC_STRIDE  (S_ + 4)     // 2052
#define KS_STRIDE  (DH_ + 1)    // 65 (produced by TDM pad-on-load)

// attention-kernel dynamic LDS layout (floats)
#define QS_OFF     0
#define SC_OFF     (16 * QS_STRIDE)                 // 1088
#define ZP_OFF     (SC_OFF + 16 * SC_STRIDE)        // 33920
#define RS_OFF     (ZP_OFF + 4 * 16 * DH_)          // 38016
#define KS_OFF     38080                            // 256-byte aligned
#define KS_CHUNK   (256 * KS_STRIDE)                // 16640 floats per buffer
#define LDS_FLOATS (KS_OFF + 2 * KS_CHUNK)          // 71360 (~279 KB)

// ---------------------------------------------------------------------------
// TDM: async DMA of one K chunk (256 rows x 64 f32, row-major, contiguous)
// into LDS with 1-dword padding after every 64 dwords (row stride -> 65).
// D# per CDNA5 ISA ch.8; groups 2/3 zero (tile dims 2+ unused for 2D).
// ---------------------------------------------------------------------------
static __device__ inline void tdm_load_k_chunk(const float* gsrc, unsigned lds_byte_off)
{
    unsigned long long ga = (unsigned long long)(const void*)gsrc;
    u32x4 g0;
    g0[0] = 1u;                                           // count=1 (valid user D#)
    g0[1] = lds_byte_off;                                 // lds_addr (bytes)
    g0[2] = (unsigned)(ga & 0xffffffffu);                 // global_addr[31:0]
    g0[3] = (unsigned)((ga >> 32) & 0x1ffffffu)           // global_addr[56:32]
          | (2u << 30);                                   // type = 2 ("image")
    i32x8 g1;
    g1[0] = (2 << 16)                                     // data_size = 4 bytes
          | (1 << 20)                                     // pad_enable
          | (5 << 22)                                     // pad_interval: every 64 dwords
          | (0 << 25);                                    // pad_amount: 1 dword
    g1[1] = (64 << 16);                                   // tensor_dim0 = 64
    g1[2] = (256 << 16);                                  // tensor_dim1 = 256
    g1[3] = (64 << 16);                                   // tile_dim0 = 64
    g1[4] = 256;                                          // tile_dim1 = 256, tile_dim2 = 0
    g1[5] = 64;                                           // tensor_dim0_stride = 64
    g1[6] = 0;
    g1[7] = 0;
    i32x4 z4 = {0, 0, 0, 0};
#if defined(__clang_major__) && (__clang_major__ >= 23)
    i32x8 z8 = {0, 0, 0, 0, 0, 0, 0, 0};
    __builtin_amdgcn_tensor_load_to_lds(g0, g1, z4, z4, z8, 0);
#else
    __builtin_amdgcn_tensor_load_to_lds(g0, g1, z4, z4, 0);
#endif
}

// ---------------------------------------------------------------------------
// Kernel 1: fused QKV projection.
// grid: (B*S/16, H), block: 128 threads (4 wave32).
// x: [B*S, DIN]; wq/wk/wv: [H, DIN, DH]; q/k/v out: [B, H, S, DH]
// ---------------------------------------------------------------------------
__global__ __launch_bounds__(128)
void qkv_proj_kernel(const float* __restrict__ x,
                     const float* __restrict__ wq,
                     const float* __restrict__ wk,
                     const float* __restrict__ wv,
                     float* __restrict__ q,
                     float* __restrict__ k,
                     float* __restrict__ v)
{
    __shared__ float xs[16 * XS_STRIDE];       // ~48 KB padded x-tile
    const int row0 = blockIdx.x * 16;
    const int h    = blockIdx.y;
    const int tid  = threadIdx.x;
    const int lane = tid & 31;
    const int wid  = tid >> 5;

    // cooperative vectorized load, padded rows (XS_STRIDE % 4 == 0)
    const float4* xsrc = (const float4*)(x + (size_t)row0 * DIN_);
    for (int i = tid; i < 16 * (DIN_ / 4); i += blockDim.x) {
        const int r = i / (DIN_ / 4);
        const int c4 = i % (DIN_ / 4);
        *(float4*)(xs + r * XS_STRIDE + c4 * 4) = xsrc[r * (DIN_ / 4) + c4];
    }
    __syncthreads();

    const int n0   = wid * 16;
    const int m    = lane & 15;                // A-matrix row (ISA 7.12.2)
    const int nlan = lane & 15;                // B/C/D column
    const int ksel = (lane >> 4) << 1;         // lanes 16-31 hold K+2,K+3
    const int half = (lane >> 4) * 8;          // C/D: lanes 16-31 hold M+8

    v8f cq = {}; v8f ck = {}; v8f cv = {};
    const float* wqh = wq + (size_t)h * DIN_ * DH_;
    const float* wkh = wk + (size_t)h * DIN_ * DH_;
    const float* wvh = wv + (size_t)h * DIN_ * DH_;

    for (int kk = 0; kk < DIN_; kk += 4) {
        v2f a;
        a.x = xs[m * XS_STRIDE + kk + ksel];
        a.y = xs[m * XS_STRIDE + kk + ksel + 1];
        const int kr0 = (kk + ksel) * DH_ + n0 + nlan;
        v2f bq, bk, bv;
        bq.x = wqh[kr0]; bq.y = wqh[kr0 + DH_];
        bk.x = wkh[kr0]; bk.y = wkh[kr0 + DH_];
        bv.x = wvh[kr0]; bv.y = wvh[kr0 + DH_];
        // three independent accumulators per A-fragment hide WMMA RAW latency
        cq = __builtin_amdgcn_wmma_f32_16x16x4_f32(false, a, false, bq, (short)0, cq, false, false);
        ck = __builtin_amdgcn_wmma_f32_16x16x4_f32(false, a, false, bk, (short)0, ck, false, false);
        cv = __builtin_amdgcn_wmma_f32_16x16x4_f32(false, a, false, bv, (short)0, cv, false, false);
    }

    const int b  = row0 >> 11;
    const int s0 = row0 & (S_ - 1);
    size_t base = ((size_t)(b * H_ + h) * S_ + s0) * DH_ + n0 + nlan;
    for (int r = 0; r < 8; ++r) {
        size_t idx = base + (size_t)(r + half) * DH_;
        q[idx] = cq[r]; k[idx] = ck[r]; v[idx] = cv[r];
    }
}

// ---------------------------------------------------------------------------
// Kernel 2: attention for one (b, h, 16-query tile).
// grid: (S/16, H, B), block: 512 threads (16 wave32), dynamic LDS ~279 KB.
// K is staged through LDS by the Tensor Data Mover (double-buffered 256-key
// chunks, pad-on-load for conflict-free transposed reads); scores strip
// (16 x 2048) lives entirely in the 320 KB WGP LDS -> exact softmax.
// z out: [B, S, H, DH]
// ---------------------------------------------------------------------------
__global__ __launch_bounds__(512)
void attention_kernel(const float* __restrict__ q,
                      const float* __restrict__ k,
                      const float* __restrict__ v,
                      float* __restrict__ z)
{
    extern __shared__ float lds[];
    float* qs     = lds + QS_OFF;
    float* sc     = lds + SC_OFF;
    float* zpart  = lds + ZP_OFF;
    float* rowsum = lds + RS_OFF;

    const int qt   = blockIdx.x;
    const int h    = blockIdx.y;
    const int b    = blockIdx.z;
    const int tid  = threadIdx.x;
    const int lane = tid & 31;
    const int wid  = tid >> 5;
    const int s0   = qt * 16;

    const size_t headbase = (size_t)(b * H_ + h) * S_ * DH_;
    const float* kbase = k + headbase;

    // kick off the DMA of K chunk 0 as early as possible
    if (wid == 0)
        tdm_load_k_chunk(kbase, KS_OFF * 4u);

    // stage Q tile (16 x 64) with padded rows
    for (int i = tid; i < 16 * DH_; i += blockDim.x) {
        const int mr = i >> 6;
        const int d  = i & 63;
        qs[mr * QS_STRIDE + d] = q[headbase + (size_t)s0 * DH_ + i];
    }

    const int m    = lane & 15;
    const int nlan = lane & 15;
    const int ksel = (lane >> 4) << 1;
    const int half = (lane >> 4) * 8;

    // ---- Phase 1: scores = (Q K^T)/8; 8 chunks of 256 keys, 1 tile/wave/chunk
    for (int c = 0; c < 8; ++c) {
        if (wid == 0) {
            if (c + 1 < 8) {
                tdm_load_k_chunk(kbase + (size_t)(c + 1) * 256 * DH_,
                                 (unsigned)(KS_OFF + ((c + 1) & 1) * KS_CHUNK) * 4u);
                __builtin_amdgcn_s_wait_tensorcnt(1);   // chunk c landed (in-order)
            } else {
                __builtin_amdgcn_s_wait_tensorcnt(0);
            }
        }
        __syncthreads();

        const float* ks = lds + KS_OFF + (c & 1) * KS_CHUNK;
        const int key0 = c * 256 + wid * 16;            // global key index
        const int krow = wid * 16 + nlan;               // row within chunk
        v8f cacc = {};
        for (int kk = 0; kk < DH_; kk += 4) {
            v2f a;
            a.x = qs[m * QS_STRIDE + kk + ksel];
            a.y = qs[m * QS_STRIDE + kk + ksel + 1];
            // B[d][n] = K[key0+n, d] from padded LDS (conflict-free)
            v2f bb;
            bb.x = ks[krow * KS_STRIDE + kk + ksel];
            bb.y = ks[krow * KS_STRIDE + kk + ksel + 1];
            cacc = __builtin_amdgcn_wmma_f32_16x16x4_f32(false, a, false, bb, (short)0, cacc, false, false);
        }
        for (int r = 0; r < 8; ++r)
            sc[(r + half) * SC_STRIDE + key0 + nlan] = cacc[r] * 0.125f;
        __syncthreads();   // protect double-buffer reuse
    }

    // ---- Phase 2: exact softmax, one wave per query row (wave32 shuffles)
    {
        const int row = wid;
        float* srow = sc + row * SC_STRIDE;
        float mx = -3.0e38f;
        for (int i = lane; i < S_; i += 32) mx = fmaxf(mx, srow[i]);
        for (int off = 16; off > 0; off >>= 1) mx = fmaxf(mx, __shfl_xor(mx, off, 32));
        float sum = 0.0f;
        for (int i = lane; i < S_; i += 32) {
            float e = __expf(srow[i] - mx);
            srow[i] = e;
            sum += e;
        }
        for (int off = 16; off > 0; off >>= 1) sum += __shfl_xor(sum, off, 32);
        if (lane == 0) rowsum[row] = sum;
    }
    __syncthreads();

    // ---- Phase 3: Z = P * V; 16 waves = 4 key-chunks (512) x 4 d-tiles
    {
        const int kg = wid >> 2;
        const int nt = wid & 3;
        const int d0 = nt * 16;
        const int nbase = kg * 512;
        v8f cacc = {};
        for (int nn = 0; nn < 512; nn += 4) {
            const int kdim = nbase + nn;
            v2f a;
            a.x = sc[m * SC_STRIDE + kdim + ksel];
            a.y = sc[m * SC_STRIDE + kdim + ksel + 1];
            // B[n][d] = V[kdim+n, d0+d] (coalesced across the 16 lanes)
            const float* vp = v + headbase + (size_t)(kdim + ksel) * DH_ + d0 + nlan;
            v2f bb; bb.x = vp[0]; bb.y = vp[DH_];
            cacc = __builtin_amdgcn_wmma_f32_16x16x4_f32(false, a, false, bb, (short)0, cacc, false, false);
        }
        for (int r = 0; r < 8; ++r)
            zpart[(kg * 16 + r + half) * DH_ + d0 + nlan] = cacc[r];
    }
    __syncthreads();

    // ---- Phase 4: reduce partials, normalize, store as [B, S, H, DH]
    for (int i = tid; i < 16 * DH_; i += blockDim.x) {
        const int mr = i >> 6;
        const int d  = i & 63;
        float acc = zpart[i] + zpart[16 * DH_ + i] + zpart[2 * 16 * DH_ + i]
                  + zpart[3 * 16 * DH_ + i];
        acc *= (1.0f / rowsum[mr]);
        z[(((size_t)b * S_ + s0 + mr) * H_ + h) * DH_ + d] = acc;
    }
}

// ---------------------------------------------------------------------------
// Kernel 3: output projection. out[r, i] = sum_j wc[i, j] * z[r, j]
// grid: (B*S/16), block: 128 (4 waves x 16-wide N tiles => 64 outputs)
// ---------------------------------------------------------------------------
__global__ __launch_bounds__(128)
void out_proj_kernel(const float* __restrict__ z,   // [B*S, 768]
                     const float* __restrict__ wc,  // [64, 768]
                     float* __restrict__ out)       // [B*S, 64]
{
    __shared__ float zs[16 * XS_STRIDE];
    const int row0 = blockIdx.x * 16;
    const int tid  = threadIdx.x;
    const int lane = tid & 31;
    const int wid  = tid >> 5;

    const float4* src = (const float4*)(z + (size_t)row0 * DIN_);
    for (int i = tid; i < 16 * (DIN_ / 4); i += blockDim.x) {
        const int r = i / (DIN_ / 4);
        const int c4 = i % (DIN_ / 4);
        *(float4*)(zs + r * XS_STRIDE + c4 * 4) = src[r * (DIN_ / 4) + c4];
    }
    __syncthreads();

    const int n0   = wid * 16;
    const int m    = lane & 15;
    const int nlan = lane & 15;
    const int ksel = (lane >> 4) << 1;
    const int half = (lane >> 4) * 8;

    v8f c = {};
    for (int kk = 0; kk < DIN_; kk += 4) {
        v2f a;
        a.x = zs[m * XS_STRIDE + kk + ksel];
        a.y = zs[m * XS_STRIDE + kk + ksel + 1];
        // B[j][i] = wc[i, j]
        const float* wp = wc + (size_t)(n0 + nlan) * DIN_ + kk + ksel;
        v2f bb; bb.x = wp[0]; bb.y = wp[1];
        c = __builtin_amdgcn_wmma_f32_16x16x4_f32(false, a, false, bb, (short)0, c, false, false);
    }
    for (int r = 0; r < 8; ++r)
        out[(size_t)(row0 + r + half) * DH_ + n0 + nlan] = c[r];
}

// ---------------------------------------------------------------------------
extern "C" void kernel_launch(void* const* d_in, const int* in_sizes, int n_in,
                              void* d_out, int out_size, void* d_ws, size_t ws_size,
                              hipStream_t stream) {
    (void)in_sizes; (void)n_in; (void)out_size; (void)ws_size;
    const float* x  = (const float*)d_in[0];
    const float* wq = (const float*)d_in[1];
    const float* wk = (const float*)d_in[2];
    const float* wv = (const float*)d_in[3];
    const float* wc = (const float*)d_in[4];
    float* out = (float*)d_out;

    float* ws = (float*)d_ws;
    const size_t qsz = (size_t)B_ * H_ * S_ * DH_;
    float* q = ws;
    float* k = ws + qsz;
    float* v = ws + 2 * qsz;
    float* z = ws + 3 * qsz;                          // [B, S, H*DH]

    dim3 g1(B_ * S_ / 16, H_);
    qkv_proj_kernel<<<g1, 128, 0, stream>>>(x, wq, wk, wv, q, k, v);

    const size_t lds2 = (size_t)LDS_FLOATS * sizeof(float);   // ~279 KB
    (void)hipFuncSetAttribute((const void*)attention_kernel,
                              hipFuncAttributeMaxDynamicSharedMemorySize, (int)lds2);
    dim3 g2(S_ / 16, H_, B_);
    attention_kernel<<<g2, 512, lds2, stream>>>(q, k, v, z);

    out_proj_kernel<<<B_ * S_ / 16, 128, 0, stream>>>(z, wc, out);
}